// custom_loss_57956288692862
// MI455X (gfx1250) — compile-verified
//
#include <hip/hip_runtime.h>
#include <stdint.h>

// ---------------------------------------------------------------------------
// Problem constants (from the reference):
//   x: [16384, 1024] f32  ->  4096 groups of 4 rows; q=row0, a=row1, c=row3
//   target: [2, 4096] i32
//   out: [1 + 4096] f32  (mean loss, then per-row equality as 0/1)
// ---------------------------------------------------------------------------
#define D          1024      // feature dim
#define B_ROWS     4096      // number of (q,a,c) groups
#define TILE_ROWS  16        // rows per WMMA tile (M=N=16)
#define PIECE      128       // floats of K staged per piece per row
#define NPIECES    (D / PIECE)   // 8
#define LDS_ROW    132       // padded floats per row (bank-conflict-free)

typedef float v2f __attribute__((ext_vector_type(2)));
typedef float v8f __attribute__((ext_vector_type(8)));
typedef int   v4i __attribute__((vector_size(16)));   // matches builtin param type

typedef __attribute__((address_space(1))) v4i* gv4p;  // global v4i*
typedef __attribute__((address_space(3))) v4i* sv4p;  // LDS v4i*

// ---- CDNA5 async global->LDS copy (512B per wave-instruction) -------------
__device__ __forceinline__ void async_row_copy(const float* g, const float* lbase,
                                               int lane) {
    const float* gp = g + lane * 4;                                  // 16B/lane
    uint32_t     lo = (uint32_t)(uintptr_t)lbase + (uint32_t)(lane * 16);
#if __has_builtin(__builtin_amdgcn_global_load_async_to_lds_b128)
    __builtin_amdgcn_global_load_async_to_lds_b128(
        (gv4p)(uintptr_t)gp, (sv4p)lo, /*offset=*/0, /*cpol=*/0);
#else
    asm volatile("global_load_async_to_lds_b128 %0, %1, off"
                 :: "v"(lo), "v"(gp)
                 : "memory");
#endif
}

template <int N>
__device__ __forceinline__ void wait_async() {
#if __has_builtin(__builtin_amdgcn_s_wait_asynccnt)
    __builtin_amdgcn_s_wait_asynccnt((unsigned short)N);
#else
    asm volatile("s_wait_asynccnt %0" :: "i"(N) : "memory");
#endif
    asm volatile("" ::: "memory");   // keep LDS reads behind the wait
}

#define WMMA_F32_4(A, B, C) \
    __builtin_amdgcn_wmma_f32_16x16x4_f32(false, (A), false, (B), (short)0, (C), false, false)

// ---------------------------------------------------------------------------
// Kernel 1: one wave per 16-row tile. Async-stage q/a/c K-pieces into LDS
// (double buffered), accumulate the six Gram products with f32 WMMA,
// harvest diagonals, compute per-row loss + equality, wave-reduce loss.
// ---------------------------------------------------------------------------
__global__ void __launch_bounds__(32)
loss_kernel(const float* __restrict__ x, const int* __restrict__ target,
            float* __restrict__ out, float* __restrict__ partials) {
    __shared__ float lds[2][3][TILE_ROWS * LDS_ROW];   // ~50.7 KB
    __shared__ float diag[6][TILE_ROWS];

    const int    lane    = threadIdx.x;
    const int    tile    = blockIdx.x;
    const size_t rowbase = (size_t)tile * TILE_ROWS;

    const int sub[3] = {0, 1, 3};   // q, a, c sub-rows inside each 4-row group

    auto issue = [&](int p, int b) {
#pragma unroll
        for (int m = 0; m < 3; ++m) {
#pragma unroll
            for (int i = 0; i < TILE_ROWS; ++i) {
                const float* g = x + ((rowbase + (size_t)i) * 4 + (size_t)sub[m]) * D
                                   + (size_t)p * PIECE;
                async_row_copy(g, &lds[b][m][i * LDS_ROW], lane);
            }
        }
    };

    v8f qq = {0}, aa = {0}, cc = {0}, qa = {0}, ac = {0}, qc = {0};
    const int row  = lane & 15;
    const int koff = (lane >> 4) << 1;      // lanes 16-31 hold K+2, K+3

    issue(0, 0);
#pragma unroll 1
    for (int p = 0; p < NPIECES; ++p) {
        const int b = p & 1;
        if (p + 1 < NPIECES) {
            issue(p + 1, b ^ 1);            // prefetch next piece (48 instrs)
            wait_async<48>();               // in-order: first 48 (piece p) done
        } else {
            wait_async<0>();
        }
        const float* bq = &lds[b][0][row * LDS_ROW + koff];
        const float* ba = &lds[b][1][row * LDS_ROW + koff];
        const float* bc = &lds[b][2][row * LDS_ROW + koff];
#pragma unroll 4
        for (int c4 = 0; c4 < PIECE / 4; ++c4) {   // K advances by 4 per WMMA
            v2f fq = *(const v2f*)(bq + c4 * 4);
            v2f fa = *(const v2f*)(ba + c4 * 4);
            v2f fc = *(const v2f*)(bc + c4 * 4);
            qq = WMMA_F32_4(fq, fq, qq);
            aa = WMMA_F32_4(fa, fa, aa);
            cc = WMMA_F32_4(fc, fc, cc);
            qa = WMMA_F32_4(fq, fa, qa);
            ac = WMMA_F32_4(fa, fc, ac);
            qc = WMMA_F32_4(fq, fc, qc);
        }
    }

    // Harvest diagonals: elem i<8 -> VGPR i @ lane i ; i>=8 -> VGPR i-8 @ lane i+16
    auto dump = [&](v8f a, int j) {
#pragma unroll
        for (int r = 0; r < 8; ++r) {
            float v = a[r];
            if (lane == r)           diag[j][r]     = v;
            else if (lane == r + 24) diag[j][r + 8] = v;
        }
    };
    dump(qq, 0); dump(aa, 1); dump(cc, 2); dump(qa, 3); dump(ac, 4); dump(qc, 5);
    __syncthreads();

    float lossv = 0.0f;
    if (lane < TILE_ROWS) {
        const size_t bi = rowbase + (size_t)lane;
        float nq = fmaxf(sqrtf(diag[0][lane]), 1e-12f);
        float na = fmaxf(sqrtf(diag[1][lane]), 1e-12f);
        float nc = fmaxf(sqrtf(diag[2][lane]), 1e-12f);
        float sqa = diag[3][lane] / (nq * na);   // q.a (TEMPERATURE == 1)
        float sac = diag[4][lane] / (na * nc);   // a.c
        float sqc = diag[5][lane] / (nq * nc);   // q.c
        int t0 = target[bi];
        int t1 = target[B_ROWS + bi];
        float eqa = expf(sqa), eqc = expf(sqc), eac = expf(sac);
        float up   = (1.0f - (float)t0) * eqa + (1.0f - (float)t1) * eqc;
        float down = ((float)t0) * eqa + ((float)t1) * eqc;
        up   = fmaxf(up,   1e-8f);
        down = fmaxf(down, 1e-8f);
        lossv = logf(up + eac) - logf(down);
        int correct = (sqa > sqc) ? 1 : 0;
        out[1 + bi] = (correct == t0) ? 1.0f : 0.0f;
    }
#pragma unroll
    for (int off = 16; off > 0; off >>= 1)
        lossv += __shfl_down(lossv, off, 32);
    if (lane == 0) partials[tile] = lossv;
}

// ---------------------------------------------------------------------------
// Kernel 2: deterministic tree-reduce of 256 tile partials -> mean loss.
// ---------------------------------------------------------------------------
__global__ void __launch_bounds__(256)
reduce_kernel(const float* __restrict__ partials, float* __restrict__ out, int n) {
    __shared__ float s[256];
    int tid = threadIdx.x;
    s[tid] = (tid < n) ? partials[tid] : 0.0f;
    __syncthreads();
#pragma unroll
    for (int off = 128; off > 0; off >>= 1) {
        if (tid < off) s[tid] += s[tid + off];
        __syncthreads();
    }
    if (tid == 0) out[0] = s[0] / (float)B_ROWS;
}

extern "C" void kernel_launch(void* const* d_in, const int* in_sizes, int n_in,
                              void* d_out, int out_size, void* d_ws, size_t ws_size,
                              hipStream_t stream) {
    (void)in_sizes; (void)n_in; (void)out_size; (void)ws_size;
    const float* x      = (const float*)d_in[0];
    const int*   target = (const int*)d_in[1];
    float*       out    = (float*)d_out;
    float*       parts  = (float*)d_ws;            // 256 floats of scratch

    const int ntiles = B_ROWS / TILE_ROWS;         // 256 tiles, 1 wave each
    loss_kernel<<<ntiles, 32, 0, stream>>>(x, target, out, parts);
    reduce_kernel<<<1, 256, 0, stream>>>(parts, out, ntiles);
}